// GCN_81484119540504
// MI455X (gfx1250) — compile-verified
//
#include <hip/hip_runtime.h>

typedef __attribute__((ext_vector_type(2))) float v2f;
typedef __attribute__((ext_vector_type(8))) float v8f;

// ---------------------------------------------------------------------------
// Degree / normalization
// ---------------------------------------------------------------------------
__global__ void deg_init_kernel(float* deg, int n) {
    int i = blockIdx.x * blockDim.x + threadIdx.x;
    if (i < n) deg[i] = 1.0f;                      // self-loop contributes 1
}

__global__ void deg_edges_kernel(const int* __restrict__ dst, float* deg, int E) {
    int e = blockIdx.x * blockDim.x + threadIdx.x;
    if (e < E) atomicAdd(&deg[dst[e]], 1.0f);
}

__global__ void dinv_kernel(float* dinv, int n) {
    int i = blockIdx.x * blockDim.x + threadIdx.x;
    if (i < n) {
        float d = dinv[i];                         // holds degree
        dinv[i] = (d > 0.0f) ? rsqrtf(fmaxf(d, 1.0f)) : 0.0f;
    }
}

// ---------------------------------------------------------------------------
// T = H (N x din) @ W (din x dout)  via V_WMMA_F32_16X16X4_F32
// One wave per 16x16 output tile. din, dout multiples of 16 (here of 4 too).
// ---------------------------------------------------------------------------
__global__ __launch_bounds__(32)
void gemm_wmma_f32(const float* __restrict__ H, const float* __restrict__ W,
                   float* __restrict__ T, int din, int dout) {
    const int lane = threadIdx.x;          // 0..31, full wave
    const int m    = lane & 15;            // row within tile (A) / col within tile (B,C)
    const int half = lane >> 4;            // 0 or 1
    const int row0 = blockIdx.x * 16;
    const int col0 = blockIdx.y * 16;

    const float* hrow = H + (size_t)(row0 + m) * din;   // A: lane m = matrix row m
    const float* wcol = W + col0 + m;                   // B: lane m = matrix col m

    v8f acc = {};
    for (int k = 0; k < din; k += 4) {
        const int ka = k + 2 * half;       // lanes 0-15: K=k,k+1 ; lanes 16-31: K=k+2,k+3
        v2f a, b;
        a.x = hrow[ka];
        a.y = hrow[ka + 1];
        b.x = wcol[(size_t)ka * dout];
        b.y = wcol[(size_t)(ka + 1) * dout];
        // (neg_a, A, neg_b, B, c_mod, C, reuse_a, reuse_b)
        acc = __builtin_amdgcn_wmma_f32_16x16x4_f32(
            false, a, false, b, (short)0, acc, false, false);
    }

    // C/D: VGPR v -> row (row0 + v + 8*half), col (col0 + m)
    float* trow = T + (size_t)(row0 + 8 * half) * dout + col0 + m;
#pragma unroll
    for (int v = 0; v < 8; ++v)
        trow[(size_t)v * dout] = acc[v];
}

// Last layer: dout == 1, plain per-row dot product (tiny: 20000 x 128 MACs)
__global__ void matvec_kernel(const float* __restrict__ H, const float* __restrict__ W,
                              float* __restrict__ T, int din, int n) {
    int i = blockIdx.x * blockDim.x + threadIdx.x;
    if (i >= n) return;
    const float* h = H + (size_t)i * din;
    float s = 0.0f;
    for (int k = 0; k < din; ++k) s = fmaf(h[k], W[k], s);
    T[i] = s;
}

// ---------------------------------------------------------------------------
// Aggregation
// ---------------------------------------------------------------------------
__global__ void zero_kernel(float* p, long n) {
    long i = (long)blockIdx.x * blockDim.x + threadIdx.x;
    if (i < n) p[i] = 0.0f;
}

// One thread = 4 features of one edge (dout % 4 == 0)
__global__ void scatter_vec4_kernel(const int* __restrict__ src, const int* __restrict__ dst,
                                    const float* __restrict__ dinv, const float* __restrict__ T,
                                    float* agg, int dout, int E) {
    long idx = (long)blockIdx.x * blockDim.x + threadIdx.x;
    int chunks = dout >> 2;
    long e = idx / chunks;
    if (e >= E) return;
    int c = (int)(idx - e * chunks);
    int s = src[e], d = dst[e];
    float w = dinv[s] * dinv[d];
    const float4 v = *((const float4*)(T + (size_t)s * dout) + c);
    float* out = agg + (size_t)d * dout + (size_t)c * 4;
    atomicAdd(out + 0, v.x * w);
    atomicAdd(out + 1, v.y * w);
    atomicAdd(out + 2, v.z * w);
    atomicAdd(out + 3, v.w * w);
}

// Scalar fallback (dout == 1 last layer)
__global__ void scatter_scalar_kernel(const int* __restrict__ src, const int* __restrict__ dst,
                                      const float* __restrict__ dinv, const float* __restrict__ T,
                                      float* agg, int dout, int E) {
    long idx = (long)blockIdx.x * blockDim.x + threadIdx.x;
    long e = idx / dout;
    if (e >= E) return;
    int j = (int)(idx - e * dout);
    int s = src[e], d = dst[e];
    float w = dinv[s] * dinv[d];
    atomicAdd(&agg[(size_t)d * dout + j], T[(size_t)s * dout + j] * w);
}

// agg[i,j] = relu?( agg[i,j] + T[i,j]*dinv[i]^2 + b[j] )   (self-loop folded in)
__global__ void finalize_kernel(float* agg, const float* __restrict__ T,
                                const float* __restrict__ dinv, const float* __restrict__ bias,
                                int dout, int n, int do_relu) {
    long idx = (long)blockIdx.x * blockDim.x + threadIdx.x;
    long total = (long)n * dout;
    if (idx >= total) return;
    int i = (int)(idx / dout);
    int j = (int)(idx - (long)i * dout);
    float di = dinv[i];
    float v = agg[idx] + T[idx] * di * di + bias[j];
    if (do_relu) v = fmaxf(v, 0.0f);
    agg[idx] = v;
}

// ---------------------------------------------------------------------------
// Host launcher
// ---------------------------------------------------------------------------
extern "C" void kernel_launch(void* const* d_in, const int* in_sizes, int n_in,
                              void* d_out, int out_size, void* d_ws, size_t ws_size,
                              hipStream_t stream) {
    static const int DIMS[11] = {128, 128, 192, 256, 256, 256, 256, 256, 192, 128, 1};
    const int NL = 10;

    const float* x  = (const float*)d_in[0];
    const int*   ei = (const int*)d_in[1];
    const int E = in_sizes[1] / 2;
    const int N = in_sizes[0] / DIMS[0];
    const int* src = ei;        // edge_index[0]
    const int* dst = ei + E;    // edge_index[1]

    const float* Ws[10];
    const float* bs[10];
    for (int i = 0; i < NL; ++i) {
        Ws[i] = (const float*)d_in[2 + i];
        bs[i] = (const float*)d_in[2 + NL + i];
    }

    // Scratch layout: dinv | buf0 | buf1 | T   (each big buf = N*256 floats)
    const size_t BIG = (size_t)N * 256 * sizeof(float);
    char* ws = (char*)d_ws;
    float* dinv = (float*)ws;
    float* buf0 = (float*)(ws + ((size_t)N * sizeof(float) + 255 & ~(size_t)255));
    float* buf1 = (float*)((char*)buf0 + BIG);
    float* Tbuf = (float*)((char*)buf1 + BIG);

    const int TB = 256;
    // --- degrees -> dinv (in place) ---
    deg_init_kernel<<<(N + TB - 1) / TB, TB, 0, stream>>>(dinv, N);
    deg_edges_kernel<<<(E + TB - 1) / TB, TB, 0, stream>>>(dst, dinv, E);
    dinv_kernel<<<(N + TB - 1) / TB, TB, 0, stream>>>(dinv, N);

    const float* Hin = x;
    for (int l = 0; l < NL; ++l) {
        const int din = DIMS[l], dout = DIMS[l + 1];
        const bool last = (l == NL - 1);
        float* agg = last ? (float*)d_out : ((l & 1) ? buf1 : buf0);

        // 1) T = Hin @ W
        if ((dout & 15) == 0) {
            dim3 grid(N / 16, dout / 16);
            gemm_wmma_f32<<<grid, 32, 0, stream>>>(Hin, Ws[l], Tbuf, din, dout);
        } else {
            matvec_kernel<<<(N + TB - 1) / TB, TB, 0, stream>>>(Hin, Ws[l], Tbuf, din, N);
        }

        // 2) zero aggregation target
        long nelem = (long)N * dout;
        zero_kernel<<<(unsigned)((nelem + TB - 1) / TB), TB, 0, stream>>>(agg, nelem);

        // 3) edge scatter with symmetric normalization
        if ((dout & 3) == 0) {
            long nthr = (long)E * (dout >> 2);
            scatter_vec4_kernel<<<(unsigned)((nthr + TB - 1) / TB), TB, 0, stream>>>(
                src, dst, dinv, Tbuf, agg, dout, E);
        } else {
            long nthr = (long)E * dout;
            scatter_scalar_kernel<<<(unsigned)((nthr + TB - 1) / TB), TB, 0, stream>>>(
                src, dst, dinv, Tbuf, agg, dout, E);
        }

        // 4) self-loop + bias (+ ReLU except last), in place
        finalize_kernel<<<(unsigned)((nelem + TB - 1) / TB), TB, 0, stream>>>(
            agg, Tbuf, dinv, bs[l], dout, N, last ? 0 : 1);

        Hin = agg;
    }
}